// get_loss_pre_91276644974602
// MI455X (gfx1250) — compile-verified
//
#include <hip/hip_runtime.h>

typedef __attribute__((ext_vector_type(2))) float v2f;
typedef __attribute__((ext_vector_type(8))) float v8f;

#define BATCH 8
#define N_PTS 8192
#define M_PTS 2048
#define WG_THREADS 256
#define ROWS_PER_WG 128          // 8 waves x 16 rows
#define M_CHUNKS (M_PTS / 16)
#define F_INF_BITS 0x7F800000u

__global__ void cd_init_kernel(unsigned int* __restrict__ colmin,
                               float* __restrict__ accum) {
    int i = blockIdx.x * blockDim.x + threadIdx.x;
    if (i < BATCH * M_PTS) colmin[i] = F_INF_BITS;
    if (i == 0) *accum = 0.0f;
}

__global__ __launch_bounds__(WG_THREADS)
void cd_main_kernel(const float* __restrict__ shape,     // (8, 8192, 6)
                    const float* __restrict__ skel,      // (8, 2048, 3)
                    unsigned int* __restrict__ colmin_g, // (8, 2048) sq-dist as u32
                    float* __restrict__ accum) {
    // WMMA-ready staged skeleton: per column, B-fragment halves + squared norm.
    __shared__ v2f          s_pair[M_PTS * 2];   // 32 KB  {x,y} / {z,0}
    __shared__ float        s_n2[M_PTS];         //  8 KB  x^2+y^2+z^2
    __shared__ unsigned int s_colmin[M_PTS];     //  8 KB

    const int b      = blockIdx.y;
    const int tid    = (int)threadIdx.x;
    const int wave   = tid >> 5;
    const int lane   = tid & 31;
    const int lane16 = lane & 15;
    const int khalf  = lane >> 4;   // 0: K=0,1 / rows 0-7 ; 1: K=2,3 / rows 8-15

    // One-time staging: precompute fragments + norms so the hot loop does
    // zero per-column arithmetic and zero lane-dependent selects.
    const float* skel_b = skel + (size_t)b * M_PTS * 3;
    for (int i = tid; i < M_PTS; i += WG_THREADS) {
        float x = skel_b[i * 3 + 0];
        float y = skel_b[i * 3 + 1];
        float z = skel_b[i * 3 + 2];
        v2f p0; p0.x = x; p0.y = y;
        v2f p1; p1.x = z; p1.y = 0.0f;
        s_pair[i * 2 + 0] = p0;
        s_pair[i * 2 + 1] = p1;
        s_n2[i]     = x * x + y * y + z * z;
        s_colmin[i] = F_INF_BITS;
    }
    __syncthreads();

    const int n_base = blockIdx.x * ROWS_PER_WG + wave * 16;

    // A fragment: 16x4 f32 (x, y | z, 0-pad). Row = lane&15, K-half = lane>>4.
    const float* prow = shape + ((size_t)b * N_PTS + (n_base + lane16)) * 6;
    v2f afrag;
    afrag.x = (khalf == 0) ? prow[0] : prow[2];
    afrag.y = (khalf == 0) ? prow[1] : 0.0f;

    // Squared norms of the 8 D-matrix rows this lane owns.
    float n1[8];
#pragma unroll
    for (int r = 0; r < 8; ++r) {
        const float* q = shape + ((size_t)b * N_PTS + (n_base + 8 * khalf + r)) * 6;
        n1[r] = q[0] * q[0] + q[1] * q[1] + q[2] * q[2];
    }

    float rmin[8];                     // running min of SQUARED distance per row
#pragma unroll
    for (int r = 0; r < 8; ++r) rmin[r] = 3.0e38f;

    for (int chunk = 0; chunk < M_CHUNKS; ++chunk) {
        const int col = chunk * 16 + lane16;
        v2f   bfrag = s_pair[col * 2 + khalf];   // one ds_load_b64, bank-conflict-free
        float n2    = s_n2[col];                 // broadcast read across halves

        v8f c = {};
        v8f dot = __builtin_amdgcn_wmma_f32_16x16x4_f32(
            /*neg_a=*/false, afrag, /*neg_b=*/false, bfrag,
            /*c_mod=*/(short)0, c, /*reuse_a=*/false, /*reuse_b=*/false);

        // Squared distances only; sqrt deferred past all min-reductions
        // (sqrt(max(.,0)) is monotone, so it commutes with min).
        float cmin = 3.0e38f;
#pragma unroll
        for (int r = 0; r < 8; ++r) {
            float d2 = fmaf(-2.0f, dot[r], n1[r] + n2);
            rmin[r]  = fminf(rmin[r], d2);
            cmin     = fminf(cmin, d2);
        }
        // Lanes L and L^16 hold the same column; min is associative+idempotent,
        // so both lanes issue ds_min on the same address — no shuffle, no branch.
        atomicMin(&s_colmin[col], __float_as_uint(fmaxf(cmin, 0.0f)));
    }

    // Row minima across the 16 columns held by lanes of the same K-half group.
#pragma unroll
    for (int r = 0; r < 8; ++r) {
        float v = rmin[r];
        v = fminf(v, __shfl_xor(v, 1, 32));
        v = fminf(v, __shfl_xor(v, 2, 32));
        v = fminf(v, __shfl_xor(v, 4, 32));
        v = fminf(v, __shfl_xor(v, 8, 32));
        rmin[r] = v;
    }
    if (lane16 == 0) {
        float s = 0.0f;
#pragma unroll
        for (int r = 0; r < 8; ++r)
            s += sqrtf(fmaxf(rmin[r], 0.0f));   // only 16 sqrts per wave
        atomicAdd(accum, s);                    // cd1 partial sum
    }

    __syncthreads();
    // Flush per-column sq-minima for this N-slice to global (cd2 cross-WG reduce).
    unsigned int* cm_b = colmin_g + (size_t)b * M_PTS;
    for (int i = tid; i < M_PTS; i += WG_THREADS)
        atomicMin(&cm_b[i], s_colmin[i]);
}

__global__ void cd_finalize_kernel(const unsigned int* __restrict__ colmin,
                                   const float* __restrict__ accum,
                                   float* __restrict__ out) {
    __shared__ float red[256];
    float s = 0.0f;
    for (int i = (int)threadIdx.x; i < BATCH * M_PTS; i += 256)
        s += sqrtf(__uint_as_float(colmin[i]));   // sqrt applied post-min
    red[threadIdx.x] = s;
    __syncthreads();
    for (int off = 128; off > 0; off >>= 1) {
        if ((int)threadIdx.x < off) red[threadIdx.x] += red[threadIdx.x + off];
        __syncthreads();
    }
    if (threadIdx.x == 0) out[0] = (red[0] + accum[0]) * 1.0e-4f;
}

extern "C" void kernel_launch(void* const* d_in, const int* in_sizes, int n_in,
                              void* d_out, int out_size, void* d_ws, size_t ws_size,
                              hipStream_t stream) {
    const float* shape = (const float*)d_in[0];  // (8, 8192, 6) f32
    const float* skel  = (const float*)d_in[1];  // (8, 2048, 3) f32

    unsigned int* colmin = (unsigned int*)d_ws;                       // 64 KB
    float* accum = (float*)((char*)d_ws + BATCH * M_PTS * sizeof(unsigned int));

    cd_init_kernel<<<(BATCH * M_PTS + 255) / 256, 256, 0, stream>>>(colmin, accum);

    dim3 grid(N_PTS / ROWS_PER_WG, BATCH);   // (64, 8)
    cd_main_kernel<<<grid, WG_THREADS, 0, stream>>>(shape, skel, colmin, accum);

    cd_finalize_kernel<<<1, 256, 0, stream>>>(colmin, accum, (float*)d_out);
}